// asym_tree_gen_62749472194920
// MI455X (gfx1250) — compile-verified
//
#include <hip/hip_runtime.h>
#include <math.h>

// Asymmetric binary tree cascade, DEPTH=8: T=511 nodes, L=256 leaf inputs,
// BATCH=32768. Derived closed-form wiring (verified against the Python index
// construction):
//   level 0 : leaf i (node 255+i): z = w0[i]*x[b][255-i] + b0[255+i]
//   level l : out m (of C=256>>l):  z = wl[2m]*prev[P-1-2m] + wl[2m+1]*prev[P-2-2m]
//             + bl[(C-1)+m],  P = 2C   (mirrored pairing from the 'j' counter)
//   root    : out = sigmoid(nck(z_node0)*root_w + root_b)
// Memory-bound on x (32 MB -> ~1.4us at 23.3 TB/s); compute ~50M
// transcendentals on VALU. WMMA is inapplicable (2-nnz sparse rows, 1x1 root
// GEMM) -> the CDNA5 feature used is the async global->LDS path
// (GLOBAL_LOAD_ASYNC_TO_LDS_B32 + s_wait_asynccnt) for staging the x tile.

#define NT      256     // threads per block (8 waves)
#define TB      32      // batch rows per block (== lanes per wave)
#define LX      256     // leaf inputs per row
#define XPITCH  257     // LDS row pitch for x tile: 257 % 64 == 1 -> conflict-free

struct Params {
  const float* x;
  const float* w[9];
  const float* b[9];
  const float* alpha;
  const float* beta;
  const float* gamma;
  const float* root_w;
  const float* root_b;
  float* out;
};

#if defined(__has_builtin)
#  if __has_builtin(__builtin_amdgcn_global_load_async_to_lds_b32)
#    define HAVE_ASYNC_LDS 1
#  endif
#endif
#ifndef HAVE_ASYNC_LDS
#  define HAVE_ASYNC_LDS 0
#endif

typedef __attribute__((address_space(1))) int* gptr_t;
typedef __attribute__((address_space(3))) int* lptr_t;

__device__ __forceinline__ float fast_rcp(float x) {
#if __has_builtin(__builtin_amdgcn_rcpf)
  return __builtin_amdgcn_rcpf(x);   // v_rcp_f32
#else
  return 1.0f / x;
#endif
}

// nck activation: 3 exp + 3 reciprocals + a handful of FMAs.
__device__ __forceinline__ float nck(float z, float al, float be, float ga) {
  float fna = 0.0878f * (z - 113.68f) *
              fast_rcp(1.0f + __expf(6.39f - z * 0.111358574610244989f)); // 1/8.98
  float fca = 0.129f * (z - 69.62f) *
              fast_rcp(1.0f + __expf(-4.4f - z * 0.235294117647058824f)); // 1/4.25
  float fk  = 2.23f * fast_rcp(0.436f + __expf(-0.132f * (z - 16.74f)));
  return al * fna + be * fca + ga * fk;
}

__global__ __launch_bounds__(NT) void asym_tree_kernel(Params p) {
  // Buffer A: activations [node][batch], pitch 32 (lane-consecutive accesses).
  __shared__ float sA[256 * 32];            // 32 KB
  // Buffer X: x tile [row][col] pitch 257; reused as ping buffer (needs 128*32
  // = 4096 floats <= 32*257 = 8224) from level 1 onward.
  __shared__ float sX[TB * XPITCH];         // ~32.1 KB

  const int t    = threadIdx.x;
  const int lane = t & 31;                  // == batch row within tile
  const int b0   = blockIdx.x * TB;

  // ---- stage x tile (32 rows x 256 cols) into LDS via async global->LDS ----
  {
    const float* gx = p.x + (size_t)b0 * LX;
#pragma unroll
    for (int k = 0; k < (TB * LX) / NT; ++k) {
      int wIdx = k * NT + t;                // coalesced across the block
      int bb = wIdx >> 8;
      int cc = wIdx & 255;
#if HAVE_ASYNC_LDS
      int* gsrc = (int*)(gx + wIdx);        // reinterpret (generic)
      int* ldst = (int*)(&sX[bb * XPITCH + cc]);
      __builtin_amdgcn_global_load_async_to_lds_b32(
          (gptr_t)gsrc, (lptr_t)ldst, 0, 0);
#else
      sX[bb * XPITCH + cc] = gx[wIdx];
#endif
    }
  }

  // wave-uniform scalars (L2-resident)
  const float al = *p.alpha, be = *p.beta, ga = *p.gamma;
  const float rw = *p.root_w, rb = *p.root_b;

#if HAVE_ASYNC_LDS
#  if __has_builtin(__builtin_amdgcn_s_wait_asynccnt)
  __builtin_amdgcn_s_wait_asynccnt(0);
#  else
  asm volatile("s_wait_asynccnt 0x0" ::: "memory");
#  endif
#endif
  __syncthreads();

  // ---- level 0: 256 leaves x 32 batch rows ----
  {
    const float* w0  = p.w[0];
    const float* bb0 = p.b[0];
#pragma unroll 4
    for (int k = 0; k < 32; ++k) {
      int i = (t >> 5) + k * 8;                       // wave-uniform node id
      float xv = sX[lane * XPITCH + (255 - i)];       // banks (lane+c)%64: clean
      float z  = w0[i] * xv + bb0[255 + i];
      sA[i * 32 + lane] = nck(z, al, be, ga);
    }
  }
  __syncthreads();

  // ---- levels 1..8: mirrored pairwise combine, ping-pong LDS ----
  const float* Aprev = sA;
  float*       Anext = sX;
  for (int l = 1; l <= 8; ++l) {
    const int C     = 256 >> l;     // outputs this level
    const int P     = 512 >> l;     // inputs  this level
    const int first = C - 1;        // first node id this level
    const float* wl = p.w[l];
    const float* bl = p.b[l];
    if (l < 8) {
      const int items = C * 32;
      for (int item = t; item < items; item += NT) {  // lane == item & 31
        int m   = item >> 5;
        float h1 = Aprev[(P - 1 - 2 * m) * 32 + lane];
        float h2 = Aprev[(P - 2 - 2 * m) * 32 + lane];
        float z  = wl[2 * m] * h1 + wl[2 * m + 1] * h2 + bl[first + m];
        Anext[m * 32 + lane] = nck(z, al, be, ga);
      }
      __syncthreads();
      const float* tmp = Aprev; Aprev = Anext; Anext = (float*)tmp;
    } else {
      // root: C == 1, P == 2
      if (t < 32) {
        float h1 = Aprev[32 + lane];
        float h2 = Aprev[lane];
        float z  = wl[0] * h1 + wl[1] * h2 + bl[0];
        float a0 = nck(z, al, be, ga);
        float v  = a0 * rw + rb;
        p.out[b0 + lane] = fast_rcp(1.0f + __expf(-v));  // sigmoid
      }
    }
  }
}

extern "C" void kernel_launch(void* const* d_in, const int* in_sizes, int n_in,
                              void* d_out, int out_size, void* d_ws, size_t ws_size,
                              hipStream_t stream) {
  (void)n_in; (void)d_ws; (void)ws_size; (void)out_size;
  // setup_inputs() dict order: x, (w0,b0), (w1,b1), ..., (w8,b8),
  //                            alpha, beta, gamma, root_w, root_b
  Params p;
  p.x = (const float*)d_in[0];
  for (int l = 0; l < 9; ++l) {
    p.w[l] = (const float*)d_in[1 + 2 * l];
    p.b[l] = (const float*)d_in[2 + 2 * l];
  }
  p.alpha  = (const float*)d_in[19];
  p.beta   = (const float*)d_in[20];
  p.gamma  = (const float*)d_in[21];
  p.root_w = (const float*)d_in[22];
  p.root_b = (const float*)d_in[23];
  p.out    = (float*)d_out;

  const int B = in_sizes[0] / LX;           // 32768
  dim3 grid((B + TB - 1) / TB);             // 1024 blocks
  dim3 block(NT);
  hipLaunchKernelGGL(asym_tree_kernel, grid, block, 0, stream, p);
}